// AI4DEM_81441169867151
// MI455X (gfx1250) — compile-verified
//
#include <hip/hip_runtime.h>
#include <cmath>

// ---------------- problem geometry ----------------
#define DDIM   128
#define DMASK  127
// tile: 16 x-cells (2 per thread) x 4 x 4 ; 128 threads = 4 wave32
#define CX 16
#define TY 4
#define TZ 4
#define THX 8                 // threads along x (each handles 2 adjacent cells)
#define NTHREADS (THX * TY * TZ)   // 128
// halo extents (+-2 each side)
#define HX 20
#define HY 8
#define HZ 8
#define HXP 21                 // padded LDS row stride (bank-conflict-free)
#define HN  (HX * HY * HZ)     // 1280 real cells per slab
#define HNP (HXP * HY * HZ)    // 1344 padded slots per slab
#define NSLAB 12               // {x,y,z,vx,vy,vz} x {layer0, layer1}

struct DemParams {
  const float* src[NSLAB];  // x0,x1,y0,y1,z0,z1,vx0,vx1,vy0,vy1,vz0,vz1
  const float* mask_n;      // mask for the layer being updated
  float* out_vx;
  float* out_vy;
  float* out_vz;
  float eta;                // damping coefficient ETA
  float dt_pm;              // DT / PM
  float grav_pm;            // 9.8 * PM
  int layer;                // layer n updated by this pass
};

// Boundary spring-damper; constants mirror reference double->f32 narrowing + op order.
__device__ __forceinline__ float boundary_force(float pc, float v, float m, float eta) {
  const float LO_HI = (float)(1.5 * 0.1);                      // 0.15000000596...
  const float HI_TH = (float)(128.0 * 0.1 - 0.5 * 0.1 - 0.1);  // 12.649999...
  float lo = (pc > 0.1f && pc < LO_HI) ? m : 0.0f;
  float hi = (pc > HI_TH) ? m : 0.0f;
  return 6.0e6f * lo * (LO_HI - pc)
       - 6.0e6f * hi * (((pc - 12.8f) + 0.1f) + 0.05f)
       - eta * v * lo - eta * v * hi;
}

// one spring-damper interaction, accumulated into (fx,fy,fz)
__device__ __forceinline__ void interact(float xn, float yn, float zn,
                                         float vxn, float vyn, float vzn,
                                         float xo, float yo, float zo,
                                         float vxo, float vyo, float vzo,
                                         float eta,
                                         float& fx, float& fy, float& fz) {
  const float ddx = xn - xo;
  const float ddy = yn - yo;
  const float ddz = zn - zo;
  const float d2 = ddx * ddx + ddy * ddy + ddz * ddz;
  const float dist = sqrtf(fmaxf(d2, 1e-12f));
  const float dc = fmaxf(dist, 1e-4f);
  const float inv = 1.0f / dc;                       // single IEEE divide
  const float dvn = ((vxn - vxo) * ddx + (vyn - vyo) * ddy + (vzn - vzo) * ddz) * inv;
  const float coef =
      (dist < 0.1f) ? (6.0e6f * (dist - 0.1f) + eta * dvn) * inv : 0.0f;
  fx += coef * ddx;
  fy += coef * ddy;
  fz += coef * ddz;
}

extern "C" __global__ void __launch_bounds__(NTHREADS)
dem_collision_layer(DemParams p) {
  __shared__ float smem[NSLAB * HNP];   // 64,512 B

  const int tx = threadIdx.x;           // 0..7, handles cells 2tx, 2tx+1
  const int ty = threadIdx.y;           // 0..3
  const int tz = threadIdx.z;           // 0..3
  const int tid = tx + THX * (ty + TY * tz);
  const int x0 = blockIdx.x * CX;
  const int y0 = blockIdx.y * TY;
  const int z0 = blockIdx.z * TZ;

  // low 32 bits of generic (flat) pointer to an LDS object == LDS byte offset
  const unsigned smem_base = (unsigned)(unsigned long long)(const void*)smem;

  // ---- stage haloed tile (12 slabs) via CDNA5 async global->LDS copies.
  // Per-lane wrapped global addresses implement jnp.roll's periodicity.
  for (int f = 0; f < NSLAB; ++f) {
    const float* sp = p.src[f];
    for (int c = tid; c < HN; c += NTHREADS) {      // exactly 10 uniform iters
      const int lx = c % HX;
      const int t2 = c / HX;
      const int ly = t2 % HY;
      const int lz = t2 / HY;
      const int gx = (x0 + lx - 2) & DMASK;
      const int gy = (y0 + ly - 2) & DMASK;
      const int gz = (z0 + lz - 2) & DMASK;
      const float* gp = sp + (((gz * DDIM) + gy) * DDIM + gx);
      const unsigned lds =
          smem_base + (unsigned)((f * HNP + (lz * HY + ly) * HXP + lx) * 4);
      asm volatile("global_load_async_to_lds_b32 %0, %1, off"
                   :: "v"(lds), "v"(gp) : "memory");
    }
  }

  // own-cell global index + float2 mask load overlaps the async copies
  const int gx0 = x0 + 2 * tx, gy0 = y0 + ty, gz0 = z0 + tz;
  const int gidx = ((gz0 * DDIM) + gy0) * DDIM + gx0;      // even x -> 8B aligned
  const float2 mkv = *(const float2*)(p.mask_n + gidx);

#if __has_builtin(__builtin_amdgcn_s_wait_asynccnt)
  __builtin_amdgcn_s_wait_asynccnt(0);
#else
  asm volatile("s_wait_asynccnt 0" ::: "memory");
#endif
  __syncthreads();

  // ---- collision: 2 layers x 5x5x5 offsets, two cells per thread (VOPD ILP)
  const int n = p.layer;
  const float eta = p.eta;
  const int own0 = ((tz + 2) * HY + (ty + 2)) * HXP + (2 * tx + 2);
  const int own1 = own0 + 1;

  const float xn0  = smem[(0  + n) * HNP + own0], xn1  = smem[(0  + n) * HNP + own1];
  const float yn0  = smem[(2  + n) * HNP + own0], yn1  = smem[(2  + n) * HNP + own1];
  const float zn0  = smem[(4  + n) * HNP + own0], zn1  = smem[(4  + n) * HNP + own1];
  const float vxn0 = smem[(6  + n) * HNP + own0], vxn1 = smem[(6  + n) * HNP + own1];
  const float vyn0 = smem[(8  + n) * HNP + own0], vyn1 = smem[(8  + n) * HNP + own1];
  const float vzn0 = smem[(10 + n) * HNP + own0], vzn1 = smem[(10 + n) * HNP + own1];

  float fx0 = 0.0f, fy0 = 0.0f, fz0 = 0.0f;
  float fx1 = 0.0f, fy1 = 0.0f, fz1 = 0.0f;

  for (int m = 0; m < 2; ++m) {
    const float* sxp  = smem + (0  + m) * HNP;
    const float* syp  = smem + (2  + m) * HNP;
    const float* szp  = smem + (4  + m) * HNP;
    const float* svxp = smem + (6  + m) * HNP;
    const float* svyp = smem + (8  + m) * HNP;
    const float* svzp = smem + (10 + m) * HNP;
    for (int dz = 0; dz < 5; ++dz) {
      for (int dy = 0; dy < 5; ++dy) {
        const int rb = ((tz + dz) * HY + (ty + dy)) * HXP + 2 * tx;
        // shared 6-wide dx window serves both cells (18 loads/cell, not 30)
        float Xr[6], Yr[6], Zr[6], Ur[6], Vr[6], Wr[6];
#pragma unroll
        for (int j = 0; j < 6; ++j) {
          Xr[j] = sxp[rb + j];
          Yr[j] = syp[rb + j];
          Zr[j] = szp[rb + j];
          Ur[j] = svxp[rb + j];
          Vr[j] = svyp[rb + j];
          Wr[j] = svzp[rb + j];
        }
#pragma unroll
        for (int dx = 0; dx < 5; ++dx) {
          interact(xn0, yn0, zn0, vxn0, vyn0, vzn0,
                   Xr[dx], Yr[dx], Zr[dx], Ur[dx], Vr[dx], Wr[dx],
                   eta, fx0, fy0, fz0);
          interact(xn1, yn1, zn1, vxn1, vyn1, vzn1,
                   Xr[dx + 1], Yr[dx + 1], Zr[dx + 1], Ur[dx + 1], Vr[dx + 1], Wr[dx + 1],
                   eta, fx1, fy1, fz1);
        }
      }
    }
  }

  const float fxb0 = boundary_force(xn0, vxn0, mkv.x, eta);
  const float fyb0 = boundary_force(yn0, vyn0, mkv.x, eta);
  const float fzb0 = boundary_force(zn0, vzn0, mkv.x, eta);
  const float fxb1 = boundary_force(xn1, vxn1, mkv.y, eta);
  const float fyb1 = boundary_force(yn1, vyn1, mkv.y, eta);
  const float fzb1 = boundary_force(zn1, vzn1, mkv.y, eta);

  float2 ovx, ovy, ovz;
  ovx.x = vxn0 + p.dt_pm * mkv.x * (-fx0 + fxb0);
  ovx.y = vxn1 + p.dt_pm * mkv.y * (-fx1 + fxb1);
  ovy.x = vyn0 + p.dt_pm * mkv.x * (-fy0 + fyb0);
  ovy.y = vyn1 + p.dt_pm * mkv.y * (-fy1 + fyb1);
  ovz.x = vzn0 + p.dt_pm * mkv.x * ((-p.grav_pm - fz0) + fzb0);
  ovz.y = vzn1 + p.dt_pm * mkv.y * ((-p.grav_pm - fz1) + fzb1);

  *(float2*)(p.out_vx + gidx) = ovx;
  *(float2*)(p.out_vy + gidx) = ovy;
  *(float2*)(p.out_vz + gidx) = ovz;
}

extern "C" void kernel_launch(void* const* d_in, const int* in_sizes, int n_in,
                              void* d_out, int out_size, void* d_ws, size_t ws_size,
                              hipStream_t stream) {
  (void)in_sizes; (void)n_in; (void)out_size; (void)d_ws; (void)ws_size;
  const int N = DDIM * DDIM * DDIM;

  const float* xg  = (const float*)d_in[0];
  const float* yg  = (const float*)d_in[1];
  const float* zg  = (const float*)d_in[2];
  const float* vxg = (const float*)d_in[3];
  const float* vyg = (const float*)d_in[4];
  const float* vzg = (const float*)d_in[5];
  const float* mk  = (const float*)d_in[6];
  float* out = (float*)d_out;

  // exact-double constants matching the reference
  const double alpha = 0.6931471805599453 / 3.141592653589793;  // -log(0.5)/pi
  const double gamma = alpha / std::sqrt(alpha * alpha + 1.0);
  const double pm = 4.0 / 3.0 * 3.1415 * (0.1 * 0.1 * 0.1) * 2700.0;
  const double eta = 2.0 * gamma * std::sqrt(6.0e6 * pm);

  DemParams p{};
  p.src[0]  = xg;       p.src[1]  = xg + N;
  p.src[2]  = yg;       p.src[3]  = yg + N;
  p.src[4]  = zg;       p.src[5]  = zg + N;
  p.src[6]  = vxg;      p.src[7]  = vxg + N;
  p.src[8]  = vyg;      p.src[9]  = vyg + N;
  p.src[10] = vzg;      p.src[11] = vzg + N;
  p.mask_n = mk;
  p.out_vx = out + 0 * N;   // (comp 0, layer 0)
  p.out_vy = out + 2 * N;   // (comp 1, layer 0)
  p.out_vz = out + 4 * N;   // (comp 2, layer 0)
  p.eta = (float)eta;
  p.dt_pm = (float)(1e-4 / pm);
  p.grav_pm = (float)(9.8 * pm);
  p.layer = 0;

  dim3 block(THX, TY, TZ);
  dim3 grid(DDIM / CX, DDIM / TY, DDIM / TZ);
  hipLaunchKernelGGL(dem_collision_layer, grid, block, 0, stream, p);

  // pass 2: layer 1 sees layer 0's *updated* velocities (from d_out)
  DemParams q = p;
  q.src[6]  = out + 0 * N;   // vx layer0 (updated)
  q.src[8]  = out + 2 * N;   // vy layer0 (updated)
  q.src[10] = out + 4 * N;   // vz layer0 (updated)
  q.mask_n = mk + N;
  q.out_vx = out + 1 * N;    // (comp 0, layer 1)
  q.out_vy = out + 3 * N;    // (comp 1, layer 1)
  q.out_vz = out + 5 * N;    // (comp 2, layer 1)
  q.layer = 1;
  hipLaunchKernelGGL(dem_collision_layer, grid, block, 0, stream, q);
}